// SoftBCSLoss_39977555591489
// MI455X (gfx1250) — compile-verified
//
#include <hip/hip_runtime.h>
#include <hip/hip_bf16.h>
#include <stdint.h>

// Problem constants (match reference)
#define NBATCH   2
#define DIMV     192
#define NVOX     (DIMV * DIMV * DIMV)   // 7,077,888 voxels per batch
#define NLAB     256                    // 64 bifs * 4 stubs
#define TILE     1024                   // voxels per block-tile: 256 thr * 4
#define NTILES   (NVOX / TILE)          // 6912 exactly, no tail
#define NBLKX    1152                   // 6 tiles per block per batch

// ---------------------------------------------------------------------------
// Kernel 0: zero the workspace bins (harness poisons d_ws with 0xAA)
// ---------------------------------------------------------------------------
__global__ void SoftBCS_zero_ws(float* __restrict__ ws) {
    int i = blockIdx.x * blockDim.x + threadIdx.x;
    if (i < NBATCH * NLAB * 2) ws[i] = 0.0f;
}

// ---------------------------------------------------------------------------
// Kernel 1: streaming sigmoid + 256-bin segment sum (per batch).
// CDNA5 async global->LDS loads (ASYNCcnt) stage the three input streams,
// double-buffered so the next tile streams from HBM while the current tile
// is consumed. Async loads complete in order, so after issuing the next
// tile's 3 loads, "s_wait_asynccnt 3" guarantees the current tile landed.
// Histogram bins are privatized per block in LDS via ds_add_f32 atomics.
// ---------------------------------------------------------------------------
__global__ __launch_bounds__(256)
void SoftBCS_hist(const float* __restrict__ pred,
                  const float* __restrict__ labmap,
                  float* __restrict__ g_sum,
                  float* __restrict__ g_cnt) {
    __shared__ float s_p0[2][TILE];
    __shared__ float s_p1[2][TILE];
    __shared__ float s_lb[2][TILE];
    __shared__ float s_sum[NLAB];
    __shared__ float s_cnt[NLAB];

    const int t = threadIdx.x;
    const int b = blockIdx.y;

    s_sum[t] = 0.0f;
    s_cnt[t] = 0.0f;
    __syncthreads();

    const float* p0b = pred   + (size_t)(b * 2 + 0) * NVOX;
    const float* p1b = pred   + (size_t)(b * 2 + 1) * NVOX;
    const float* lbb = labmap + (size_t)b * NVOX;

    // LDS byte offsets for this thread's 16B staging slots, both buffers
    // (low 32 bits of the generic shared address == LDS offset)
    const uint32_t l0[2] = {(uint32_t)(uintptr_t)(&s_p0[0][t * 4]),
                            (uint32_t)(uintptr_t)(&s_p0[1][t * 4])};
    const uint32_t l1[2] = {(uint32_t)(uintptr_t)(&s_p1[0][t * 4]),
                            (uint32_t)(uintptr_t)(&s_p1[1][t * 4])};
    const uint32_t l2[2] = {(uint32_t)(uintptr_t)(&s_lb[0][t * 4]),
                            (uint32_t)(uintptr_t)(&s_lb[1][t * 4])};

    // Issue one tile's 3 async b128 loads (16B per lane per stream)
    auto issue = [&](int tile, int buf) {
        const size_t voff = (size_t)tile * TILE + (size_t)t * 4;
        const uint64_t a0 = (uint64_t)(uintptr_t)(p0b + voff);
        const uint64_t a1 = (uint64_t)(uintptr_t)(p1b + voff);
        const uint64_t a2 = (uint64_t)(uintptr_t)(lbb + voff);
        asm volatile("global_load_async_to_lds_b128 %0, %1, off"
                     :: "v"(l0[buf]), "v"(a0) : "memory");
        asm volatile("global_load_async_to_lds_b128 %0, %1, off"
                     :: "v"(l1[buf]), "v"(a1) : "memory");
        asm volatile("global_load_async_to_lds_b128 %0, %1, off"
                     :: "v"(l2[buf]), "v"(a2) : "memory");
    };

    int tile = blockIdx.x;
    int buf  = 0;
    if (tile < NTILES) issue(tile, 0);

    for (; tile < NTILES; tile += gridDim.x) {
        const int next = tile + gridDim.x;
        if (next < NTILES) {
            issue(next, buf ^ 1);
            // <=3 outstanding -> the current tile's 3 loads have completed
            asm volatile("s_wait_asynccnt 3" ::: "memory");
        } else {
            asm volatile("s_wait_asynccnt 0" ::: "memory");
        }

        const float4 p0 = *reinterpret_cast<const float4*>(&s_p0[buf][t * 4]);
        const float4 p1 = *reinterpret_cast<const float4*>(&s_p1[buf][t * 4]);
        const float4 lb = *reinterpret_cast<const float4*>(&s_lb[buf][t * 4]);

        const float p0v[4] = {p0.x, p0.y, p0.z, p0.w};
        const float p1v[4] = {p1.x, p1.y, p1.z, p1.w};
        const float lbv[4] = {lb.x, lb.y, lb.z, lb.w};

#pragma unroll
        for (int i = 0; i < 4; ++i) {
            // softmax over 2 channels, class 1 == sigmoid(p1 - p0)
            const float d  = p0v[i] - p1v[i];
            const float fg = 1.0f / (1.0f + __expf(d));
            int l = __float2int_rn(lbv[i]);
            l &= (NLAB - 1);  // labels are already in [0, 255]
            atomicAdd(&s_sum[l], fg);     // ds_add_f32
            atomicAdd(&s_cnt[l], 1.0f);   // ds_add_f32
        }
        buf ^= 1;
    }

    __syncthreads();
    // Flush privatized bins: 256 global f32 atomics per array per block
    atomicAdd(&g_sum[b * NLAB + t], s_sum[t]);
    atomicAdd(&g_cnt[b * NLAB + t], s_cnt[t]);
}

// ---------------------------------------------------------------------------
// Kernel 2: finalize. 126 (batch, bif) pairs; masked softmin over 3 stubs.
// ---------------------------------------------------------------------------
__global__ __launch_bounds__(128)
void SoftBCS_finalize(const float* __restrict__ g_sum,
                      const float* __restrict__ g_cnt,
                      float* __restrict__ out) {
    __shared__ float r_tot[128];
    __shared__ float r_nb[128];

    const int tid = threadIdx.x;
    float tot = 0.0f, nb = 0.0f;

    if (tid < NBATCH * 63) {
        const int b   = tid / 63;
        const int bif = 1 + (tid % 63);       // drop bif 0
        const int base = b * NLAB + bif * 4;  // label = bif*4 + stub

        float mean[3];
        bool  valid[3];
        int   nvalid = 0;
#pragma unroll
        for (int s = 0; s < 3; ++s) {         // stubs 1..3 (drop stub 0)
            const float c  = g_cnt[base + 1 + s];
            const float sm = g_sum[base + 1 + s];
            valid[s] = (c >= 1.0f);           // MIN_VOXELS = 1
            mean[s]  = sm / fmaxf(c, 1.0f);
            nvalid  += valid[s] ? 1 : 0;
        }

        // softmin over valid stubs: softmax of (-mean / TEMP), TEMP = 0.2
        float logit[3];
#pragma unroll
        for (int s = 0; s < 3; ++s)
            logit[s] = valid[s] ? (-5.0f * mean[s]) : -1.0e9f;

        const float m = fmaxf(logit[0], fmaxf(logit[1], logit[2]));
        float w[3], W = 0.0f;
#pragma unroll
        for (int s = 0; s < 3; ++s) { w[s] = __expf(logit[s] - m); W += w[s]; }

        float score = 0.0f;
#pragma unroll
        for (int s = 0; s < 3; ++s)
            if (valid[s]) score += mean[s] * (w[s] / W);

        if (nvalid >= 2) {                    // MIN_STUBS = 2
            tot = 1.0f - score;
            nb  = 1.0f;
        }
    }

    r_tot[tid] = tot;
    r_nb[tid]  = nb;
    __syncthreads();
    for (int s2 = 64; s2 > 0; s2 >>= 1) {
        if (tid < s2) {
            r_tot[tid] += r_tot[tid + s2];
            r_nb[tid]  += r_nb[tid + s2];
        }
        __syncthreads();
    }
    if (tid == 0)
        out[0] = (r_nb[0] > 0.0f) ? (r_tot[0] / fmaxf(r_nb[0], 1.0f)) : 0.0f;
}

// ---------------------------------------------------------------------------
extern "C" void kernel_launch(void* const* d_in, const int* in_sizes, int n_in,
                              void* d_out, int out_size, void* d_ws, size_t ws_size,
                              hipStream_t stream) {
    const float* pred = (const float*)d_in[0];   // [2,2,192,192,192] f32
    const float* lab  = (const float*)d_in[1];   // [2,1,192,192,192] f32
    float* out = (float*)d_out;                  // scalar f32
    float* ws  = (float*)d_ws;

    float* g_sum = ws;                           // [2,256]
    float* g_cnt = ws + NBATCH * NLAB;           // [2,256]

    SoftBCS_zero_ws<<<1, 1024, 0, stream>>>(ws);

    dim3 grid(NBLKX, NBATCH);
    SoftBCS_hist<<<grid, 256, 0, stream>>>(pred, lab, g_sum, g_cnt);

    SoftBCS_finalize<<<1, 128, 0, stream>>>(g_sum, g_cnt, out);
}